// ClauseFunction_33646773797499
// MI455X (gfx1250) — compile-verified
//
#include <hip/hip_runtime.h>
#include <stdint.h>

#ifndef __has_builtin
#define __has_builtin(x) 0
#endif

#define GAMMA 1.0e-3f

constexpr int Bdim = 64;
constexpr int Gdim = 2048;
constexpr int Sdim = 32;
constexpr int Ldim = 8;

constexpr int B_PER_BLOCK = 8;        // x rows staged in LDS (8 * 8KB = 64KB)
constexpr int THREADS     = 256;      // 8 waves (wave32)
constexpr int WAVES       = THREADS / 32;

#if __has_builtin(__builtin_amdgcn_exp2f)
__device__ __forceinline__ float fast_exp2(float x) { return __builtin_amdgcn_exp2f(x); }
#else
__device__ __forceinline__ float fast_exp2(float x) { return exp2f(x); }
#endif

#if __has_builtin(__builtin_amdgcn_logf)
__device__ __forceinline__ float fast_log2(float x) { return __builtin_amdgcn_logf(x); }
#else
__device__ __forceinline__ float fast_log2(float x) { return __log2f(x); }
#endif

// 16-byte int vector type matching the builtin's expected pointee
// (diagnostic showed: "__attribute__((__vector_size__(4 * sizeof(int)))) int __device__ *")
typedef int v4i_b128 __attribute__((__vector_size__(16)));
typedef __attribute__((address_space(1))) v4i_b128* gptr_b128;   // global (AS1)
typedef __attribute__((address_space(3))) v4i_b128* lptr_b128;   // LDS (AS3)

__global__ __launch_bounds__(THREADS)
void clause_softor_softand_kernel(const float* __restrict__ x,
                                  const int*   __restrict__ I,
                                  float*       __restrict__ out)
{
    __shared__ float xs[B_PER_BLOCK * Gdim];            // 64 KB

    const int b0   = blockIdx.x * B_PER_BLOCK;
    const int g    = blockIdx.y * WAVES + (threadIdx.x >> 5);
    const int lane = threadIdx.x & 31;                   // lane == s
    const int tid  = threadIdx.x;

    const float* src = x + (size_t)b0 * Gdim;

    // ---- Stage 8 rows of x (64 KB) into LDS via gfx1250 async global->LDS ----
#if defined(__gfx1250__) && __has_builtin(__builtin_amdgcn_global_load_async_to_lds_b128)
    {
        // 64KB / (16B * 256 threads) = 16 async b128 per thread
        #pragma unroll
        for (int k = 0; k < (B_PER_BLOCK * Gdim) / (4 * THREADS); ++k) {
            const int elt = (k * THREADS + tid) * 4;     // float index (4 floats / b128)
            __builtin_amdgcn_global_load_async_to_lds_b128(
                (gptr_b128)(uintptr_t)(src + elt),
                (lptr_b128)(uint32_t)(uintptr_t)(xs + elt),
                0, 0);
        }
#if __has_builtin(__builtin_amdgcn_s_wait_asynccnt)
        __builtin_amdgcn_s_wait_asynccnt(0);
#else
        asm volatile("s_wait_asynccnt 0" ::: "memory");
#endif
    }
#else
    for (int i = tid; i < (B_PER_BLOCK * Gdim) / 4; i += THREADS) {
        reinterpret_cast<float4*>(xs)[i] = reinterpret_cast<const float4*>(src)[i];
    }
#endif
    __syncthreads();

    // ---- Load this wave's indices once: lane s -> I[g][s][0..7] (two b128) ----
    const int4* ip = reinterpret_cast<const int4*>(I + ((size_t)g * Sdim + lane) * Ldim);
    const int4 ia = ip[0];
    const int4 ib = ip[1];

    const float k1 = 1.4426950408889634f / GAMMA;  // log2(e)/gamma
    const float k2 = GAMMA * 0.6931471805599453f;  // gamma*ln(2)

    #pragma unroll
    for (int bb = 0; bb < B_PER_BLOCK; ++bb) {
        const float* row = xs + bb * Gdim;

        // 8 LDS gathers per lane
        const float v0 = row[ia.x], v1 = row[ia.y], v2 = row[ia.z], v3 = row[ia.w];
        const float v4 = row[ib.x], v5 = row[ib.y], v6 = row[ib.z], v7 = row[ib.w];

        // softand over L = soft-min, stable LSE form
        const float m01 = fminf(v0, v1), m23 = fminf(v2, v3);
        const float m45 = fminf(v4, v5), m67 = fminf(v6, v7);
        const float m   = fminf(fminf(m01, m23), fminf(m45, m67));

        float ssum = fast_exp2((m - v0) * k1) + fast_exp2((m - v1) * k1)
                   + fast_exp2((m - v2) * k1) + fast_exp2((m - v3) * k1)
                   + fast_exp2((m - v4) * k1) + fast_exp2((m - v5) * k1)
                   + fast_exp2((m - v6) * k1) + fast_exp2((m - v7) * k1);

        const float p = m - k2 * fast_log2(ssum);      // softand result for s = lane

        // softor over S = soft-max across 32 lanes (wave32 butterfly)
        float M = p;
        #pragma unroll
        for (int off = 16; off > 0; off >>= 1)
            M = fmaxf(M, __shfl_xor(M, off, 32));

        float e = fast_exp2((p - M) * k1);
        #pragma unroll
        for (int off = 16; off > 0; off >>= 1)
            e += __shfl_xor(e, off, 32);

        if (lane == 0)
            out[(size_t)(b0 + bb) * Gdim + g] = M + k2 * fast_log2(e);
    }
}

extern "C" void kernel_launch(void* const* d_in, const int* in_sizes, int n_in,
                              void* d_out, int out_size, void* d_ws, size_t ws_size,
                              hipStream_t stream) {
    const float* x = (const float*)d_in[0];
    const int*   I = (const int*)d_in[1];
    float*     out = (float*)d_out;

    dim3 grid(Bdim / B_PER_BLOCK, Gdim / WAVES);   // (8, 256) blocks
    clause_softor_softand_kernel<<<grid, THREADS, 0, stream>>>(x, I, out);
}